// ProductGraphGNN_66752381714624
// MI455X (gfx1250) — compile-verified
//
#include <hip/hip_runtime.h>
#include <hip/hip_bf16.h>

typedef float v2f __attribute__((ext_vector_type(2)));
typedef float v8f __attribute__((ext_vector_type(8)));

#define N_NODES 153600
#define HID     128
#define N_TURB  100
#define ISL     48
#define OSL     12
#define B_GRAPHS 32

// ---------------- CSR construction ----------------

__global__ void zero_ints(int* p, int n) {
    int i = blockIdx.x * blockDim.x + threadIdx.x;
    if (i < n) p[i] = 0;
}

__global__ void count_edges(const long long* __restrict__ dst, int* __restrict__ cnt, int E) {
    int e = blockIdx.x * blockDim.x + threadIdx.x;
    if (e < E) atomicAdd(&cnt[(int)dst[e]], 1);
}

// per-256-block inclusive scan -> exclusive base, block sums out
__global__ void scan_blocks(const int* __restrict__ cnt, int* __restrict__ base,
                            int* __restrict__ bsum) {
    __shared__ int sh[256];
    int t = threadIdx.x;
    int i = blockIdx.x * 256 + t;
    int c = cnt[i];
    sh[t] = c;
    __syncthreads();
    #pragma unroll
    for (int off = 1; off < 256; off <<= 1) {
        int v = (t >= off) ? sh[t - off] : 0;
        __syncthreads();
        sh[t] += v;
        __syncthreads();
    }
    base[i] = sh[t] - c;              // exclusive within block
    if (t == 255) bsum[blockIdx.x] = sh[255];
}

__global__ void scan_bsums(int* bsum, int nblk) {
    if (threadIdx.x == 0 && blockIdx.x == 0) {
        int run = 0;
        for (int b = 0; b < nblk; ++b) { int v = bsum[b]; bsum[b] = run; run += v; }
    }
}

// base += block offset; also dinv[i] = rsqrt(count+1)  (self-loop included)
__global__ void finish_base_dinv(int* __restrict__ base, const int* __restrict__ cnt,
                                 const int* __restrict__ bsum, float* __restrict__ dinv) {
    int i = blockIdx.x * blockDim.x + threadIdx.x;
    if (i < N_NODES) {
        base[i] += bsum[i >> 8];
        dinv[i] = rsqrtf((float)cnt[i] + 1.0f);
    }
}

__global__ void fill_csr(const long long* __restrict__ src, const long long* __restrict__ dst,
                         const int* __restrict__ base, int* __restrict__ cursor,
                         const float* __restrict__ dinv,
                         int* __restrict__ csr_src, float* __restrict__ csr_w, int E) {
    int e = blockIdx.x * blockDim.x + threadIdx.x;
    if (e >= E) return;
    int s = (int)src[e], d = (int)dst[e];
    int p = base[d] + atomicAdd(&cursor[d], 1);
    csr_src[p] = s;
    csr_w[p]   = dinv[s] * dinv[d];
}

// ---------------- GEMM: tmp = (relu?)in @ W  via V_WMMA_F32_16X16X4_F32 ----------------
// block = 256 threads (8 waves); block computes rows [16b,16b+16) x all 128 cols.
// `in` and `tmp` may alias (tile staged in LDS before any store).

template <int K, bool RELU>
__global__ void gcn_gemm(const float* in, const float* __restrict__ W, float* tmp) {
    __shared__ float As[16][K + 4];
    const int t    = threadIdx.x;
    const int row0 = blockIdx.x * 16;

    for (int i = t; i < 16 * K; i += 256) {
        int r = i / K, c = i % K;
        float v = in[(size_t)(row0 + r) * K + c];
        if (RELU) v = fmaxf(v, 0.0f);
        As[r][c] = v;
    }
    __syncthreads();

    const int wave = t >> 5;
    const int lane = t & 31;
    const int n0   = wave * 16;
    const int nn   = n0 + (lane & 15);
    const int m    = lane & 15;
    const int kh   = (lane < 16) ? 0 : 2;   // K-half select per ISA layout

    v8f c;
    #pragma unroll
    for (int i = 0; i < 8; ++i) c[i] = 0.0f;

    #pragma unroll 4
    for (int k0 = 0; k0 < K; k0 += 4) {
        v2f a, b;
        a[0] = As[m][k0 + kh];
        a[1] = As[m][k0 + kh + 1];
        b[0] = W[(size_t)(k0 + kh) * HID + nn];
        b[1] = W[(size_t)(k0 + kh + 1) * HID + nn];
        c = __builtin_amdgcn_wmma_f32_16x16x4_f32(
                /*neg_a=*/false, a, /*neg_b=*/false, b,
                /*c_mod=*/(short)0, c, /*reuse_a=*/false, /*reuse_b=*/false);
    }

    const int rbase = (lane < 16) ? 0 : 8;
    #pragma unroll
    for (int r = 0; r < 8; ++r) {
        int grow = row0 + r + rbase;
        tmp[(size_t)grow * HID + nn] = c[r];
    }
}

// ---------------- Aggregation: acc[i] = bias + dinv[i]^2*tmp[i] + sum_j w_j*tmp[src_j] ----------------
// one wave per node; 4 channels per lane; neighbor (src,w) broadcast via shuffles.

__global__ void gcn_aggregate(const float* __restrict__ tmp,
                              const int* __restrict__ base, const int* __restrict__ cnt,
                              const int* __restrict__ csr_src, const float* __restrict__ csr_w,
                              const float* __restrict__ dinv, const float* __restrict__ bias,
                              float* __restrict__ acc) {
    int wid  = (blockIdx.x * blockDim.x + threadIdx.x) >> 5;
    int lane = threadIdx.x & 31;
    if (wid >= N_NODES) return;

    const size_t row = (size_t)wid * HID;
    float dv  = dinv[wid];
    float dv2 = dv * dv;
    float r0 = bias[lane]      + dv2 * tmp[row + lane];
    float r1 = bias[lane + 32] + dv2 * tmp[row + lane + 32];
    float r2 = bias[lane + 64] + dv2 * tmp[row + lane + 64];
    float r3 = bias[lane + 96] + dv2 * tmp[row + lane + 96];

    int b0i = base[wid];
    int n   = cnt[wid];
    for (int jj = 0; jj < n; jj += 32) {
        int   sj = 0;
        float wj = 0.0f;
        if (jj + lane < n) { sj = csr_src[b0i + jj + lane]; wj = csr_w[b0i + jj + lane]; }
        int mcnt = n - jj; if (mcnt > 32) mcnt = 32;
        for (int u = 0; u < mcnt; ++u) {
            int   sb = __shfl(sj, u, 32);
            float wb = __shfl(wj, u, 32);
            const float* tr = tmp + (size_t)sb * HID;
            r0 += wb * tr[lane];
            r1 += wb * tr[lane + 32];
            r2 += wb * tr[lane + 64];
            r3 += wb * tr[lane + 96];
        }
    }
    acc[row + lane]      = r0;
    acc[row + lane + 32] = r1;
    acc[row + lane + 64] = r2;
    acc[row + lane + 96] = r3;
}

// ---------------- Head: pred = relu(h)[idx] @ Wh + bh ----------------

__global__ void gcn_head(const float* __restrict__ h, const float* __restrict__ Wh,
                         const float* __restrict__ bh, float* __restrict__ out) {
    int i = blockIdx.x * blockDim.x + threadIdx.x;     // 3200*12
    if (i >= B_GRAPHS * N_TURB * OSL) return;
    int r = i / OSL, o = i % OSL;
    int g = r / N_TURB, tb = r % N_TURB;
    size_t node = (size_t)g * (N_TURB * ISL) + (ISL - 1) * N_TURB + tb;
    const float* hr = h + node * HID;
    float s = bh[o];
    #pragma unroll 8
    for (int k = 0; k < HID; ++k) s += fmaxf(hr[k], 0.0f) * Wh[k * OSL + o];
    out[i] = s;
}

// ---------------- launch ----------------

extern "C" void kernel_launch(void* const* d_in, const int* in_sizes, int n_in,
                              void* d_out, int out_size, void* d_ws, size_t ws_size,
                              hipStream_t stream) {
    const float*     x   = (const float*)d_in[0];
    const long long* ei  = (const long long*)d_in[1];        // (2, E) int64
    const float*     W0  = (const float*)d_in[4];
    const float*     b0  = (const float*)d_in[5];
    const float*     W1  = (const float*)d_in[6];
    const float*     b1  = (const float*)d_in[7];
    const float*     W2  = (const float*)d_in[8];
    const float*     b2  = (const float*)d_in[9];
    const float*     Wh  = (const float*)d_in[10];
    const float*     bh  = (const float*)d_in[11];
    float* out = (float*)d_out;

    const int E = in_sizes[1] / 2;
    const long long* src = ei;
    const long long* dst = ei + E;

    // workspace layout
    char* p = (char*)d_ws;
    const size_t NB = (size_t)N_NODES * HID;
    float* A      = (float*)p; p += NB * sizeof(float);
    float* B      = (float*)p; p += NB * sizeof(float);
    float* dinv   = (float*)p; p += (size_t)N_NODES * sizeof(float);
    int*   cnt    = (int*)p;   p += (size_t)N_NODES * sizeof(int);
    int*   cursor = (int*)p;   p += (size_t)N_NODES * sizeof(int);
    int*   base   = (int*)p;   p += (size_t)N_NODES * sizeof(int);
    int*   bsum   = (int*)p;   p += 1024 * sizeof(int);
    int*   csrS   = (int*)p;   p += (size_t)E * sizeof(int);
    float* csrW   = (float*)p;

    const int TB = 256;
    const int nScanBlk = N_NODES / 256;                  // 600

    // CSR build (count + cursor share one zeroing pass; they're adjacent)
    zero_ints<<<(2 * N_NODES + TB - 1) / TB, TB, 0, stream>>>(cnt, 2 * N_NODES);
    count_edges<<<(E + TB - 1) / TB, TB, 0, stream>>>(dst, cnt, E);
    scan_blocks<<<nScanBlk, 256, 0, stream>>>(cnt, base, bsum);
    scan_bsums<<<1, 32, 0, stream>>>(bsum, nScanBlk);
    finish_base_dinv<<<(N_NODES + TB - 1) / TB, TB, 0, stream>>>(base, cnt, bsum, dinv);
    fill_csr<<<(E + TB - 1) / TB, TB, 0, stream>>>(src, dst, base, cursor, dinv, csrS, csrW, E);

    const int gGemm = N_NODES / 16;                      // 9600 blocks
    const int gAgg  = (N_NODES * 32 + TB - 1) / TB;      // 19200 blocks (wave/node)

    // Layer 0: tmp(A) = x @ W0 ; acc(B) = b0 + norm-aggregate
    gcn_gemm<16, false><<<gGemm, TB, 0, stream>>>(x, W0, A);
    gcn_aggregate<<<gAgg, TB, 0, stream>>>(A, base, cnt, csrS, csrW, dinv, b0, B);

    // Layer 1: tmp = relu(B) @ W1 in-place into B ; acc -> A
    gcn_gemm<128, true><<<gGemm, TB, 0, stream>>>(B, W1, B);
    gcn_aggregate<<<gAgg, TB, 0, stream>>>(B, base, cnt, csrS, csrW, dinv, b1, A);

    // Layer 2: tmp = relu(A) @ W2 in-place into A ; acc -> B
    gcn_gemm<128, true><<<gGemm, TB, 0, stream>>>(A, W2, A);
    gcn_aggregate<<<gAgg, TB, 0, stream>>>(A, base, cnt, csrS, csrW, dinv, b2, B);

    // Head
    const int nPred = B_GRAPHS * N_TURB * OSL;           // 38400
    gcn_head<<<(nPred + TB - 1) / TB, TB, 0, stream>>>(B, Wh, bh, out);
}